// CombinePatches_11879879543950
// MI455X (gfx1250) — compile-verified
//
#include <hip/hip_runtime.h>

typedef __attribute__((ext_vector_type(4))) float        v4f;
typedef __attribute__((ext_vector_type(4))) unsigned int v4u;
typedef __attribute__((ext_vector_type(8))) unsigned int v8u;

// Problem constants from the reference (B,D,H,W,C)=(2,64,128,128,4), K=4, S=2.
#define D_  64
#define H_  128
#define W_  128
#define C_  4
#define OD_ 31
#define OH_ 63
#define OW_ 63
#define PR_ 256                 // floats per patch row: 4*4*4*C
#define SEG_ 16                 // floats per (i,j) segment: 4 l-values * C
#define TILE_FLOATS_ (OW_ * SEG_)   // 63*16 = 1008 floats = 4032 B per combo tile

__global__ __launch_bounds__(128)
void fold_tdm_kernel(const float* __restrict__ patches,
                     float* __restrict__ out) {
    constexpr long SOH = (long)OW_ * PR_;           // oh stride (floats)
    constexpr long SOD = (long)OH_ * OW_ * PR_;     // od stride
    constexpr long SB  = (long)OD_ * OH_ * OW_ * PR_;

    __shared__ __align__(128) float lds_tiles[4 * TILE_FLOATS_];  // 16128 B

    const int w = threadIdx.x;   // 0..127
    const int h = blockIdx.x;    // 0..127
    const int d = blockIdx.y;    // 0..63
    const int b = blockIdx.z;    // 0..1

    // Block-uniform contribution lists for d and h dimensions.
    int ksd[2], osd[2], nd = 0;
    {
        const int o0 = d >> 1, k0 = d & 1;
        if (o0 <= OD_ - 1) { ksd[nd] = k0;     osd[nd] = o0;     ++nd; }
        if (o0 >= 1)       { ksd[nd] = k0 + 2; osd[nd] = o0 - 1; ++nd; }
    }
    int ksh[2], osh[2], nh = 0;
    {
        const int o0 = h >> 1, k0 = h & 1;
        if (o0 <= OH_ - 1) { ksh[nh] = k0;     osh[nh] = o0;     ++nh; }
        if (o0 >= 1)       { ksh[nh] = k0 + 2; osh[nh] = o0 - 1; ++nh; }
    }
    const int ncombo = nd * nh;                 // 1, 2 or 4

    const float* __restrict__ pb = patches + (long)b * SB;

    // ---- TDM stage: wave wv DMAs combo wv's 63x16-float strided tile into LDS.
    const int wv = threadIdx.x >> 5;            // wave32 id, 0..3
    if (wv < ncombo) {
        const int a = wv / nh, e = wv % nh;     // (od,oh) combo for this wave
        const float* gsrc = pb + (long)osd[a] * SOD + (long)osh[e] * SOH
                          + (ksd[a] * 64 + ksh[e] * SEG_);
        const unsigned long long gaddr = (unsigned long long)(const void*)gsrc;
        const unsigned lds_off =
            (unsigned)(unsigned long long)(const void*)&lds_tiles[wv * TILE_FLOATS_];

        // Tensor DMA Descriptor (D#), 2D tensor -> groups 0 and 1 only.
        v4u g0;
        g0.x = 1u;                                         // count=1, valid D#
        g0.y = lds_off;                                    // lds_addr (bytes)
        g0.z = (unsigned)gaddr;                            // global_addr[31:0]
        g0.w = (unsigned)((gaddr >> 32) & 0x01FFFFFFull)   // global_addr[56:32]
             | (2u << 30);                                 // type = 2 ("image")
        v8u g1;
        g1[0] = (2u << 16);          // workgroup_mask=0, data_size=2 (4 bytes)
        g1[1] = ((unsigned)SEG_ << 16);   // tensor_dim0[15:0] = 16
        g1[2] = ((unsigned)OW_  << 16);   // tensor_dim0[31:16]=0 | tensor_dim1[15:0] = 63
        g1[3] = ((unsigned)SEG_ << 16);   // tensor_dim1[31:16]=0 | tile_dim0 = 16
        g1[4] = (unsigned)OW_;            // tile_dim1 = 63 | tile_dim2 = 0
        g1[5] = (unsigned)PR_;            // tensor_dim0_stride[31:0] = 256 elements
        g1[6] = 0u;                       // dim0_stride[47:32]=0 | dim1_stride[15:0]=0
        g1[7] = 0u;                       // tensor_dim1_stride upper bits = 0

        // 2D tensor: VADDR2/VADDR3 omitted (NULL). Tracked by TENSORcnt.
        asm volatile("tensor_load_to_lds %0, %1" :: "s"(g0), "s"(g1) : "memory");
        __builtin_amdgcn_s_wait_tensorcnt(0);
    }
    __syncthreads();

    // ---- Compute stage: per-thread w-dimension contribution list.
    int ksw[2], osw[2], nw = 0;
    {
        const int o0 = w >> 1, k0 = w & 1;
        if (o0 <= OW_ - 1) { ksw[nw] = k0;     osw[nw] = o0;     ++nw; }
        if (o0 >= 1)       { ksw[nw] = k0 + 2; osw[nw] = o0 - 1; ++nw; }
    }

    v4f acc = 0.0f;
    for (int c = 0; c < ncombo; ++c) {
        const float* tile = &lds_tiles[c * TILE_FLOATS_];
        for (int f = 0; f < nw; ++f)
            acc += *(const v4f*)(tile + osw[f] * SEG_ + ksw[f] * 4);
    }

    // cnt in {1,2,4,8} -> 1/cnt exact in fp32.
    acc *= (1.0f / (float)(ncombo * nw));

    const long oidx = ((((long)b * D_ + d) * H_ + h) * (long)W_ + w) * C_;
    // Output is write-once: NT store keeps the 192MB L2 warm for patch tiles.
    __builtin_nontemporal_store(acc, (v4f*)(out + oidx));
}

extern "C" void kernel_launch(void* const* d_in, const int* in_sizes, int n_in,
                              void* d_out, int out_size, void* d_ws, size_t ws_size,
                              hipStream_t stream) {
    (void)in_sizes; (void)n_in; (void)d_ws; (void)ws_size; (void)out_size;
    const float* patches = (const float*)d_in[0];   // [2,31,63,63,256] f32
    float* out = (float*)d_out;                     // [2,64,128,128,4] f32

    dim3 block(W_);            // 128 threads = 4 wave32s, one full W row
    dim3 grid(H_, D_, 2);      // (h, d, b)
    fold_tdm_kernel<<<grid, block, 0, stream>>>(patches, out);
}